// GATLayer_40724879901269
// MI455X (gfx1250) — compile-verified
//
#include <hip/hip_runtime.h>
#include <hip/hip_bf16.h>

#define N_NODES   50000
#define N_EDGES   800000
#define DIM       128
#define NEG_SLOPE 0.01f
#define APAD      132   // A-tile row stride in LDS (floats); 132%64=4 -> conflict-free

typedef float v2f __attribute__((ext_vector_type(2)));
typedef float v8f __attribute__((ext_vector_type(8)));

// ---------------------------------------------------------------------------
// Phase 0: zero outputs / denom, init running max to -inf
// ---------------------------------------------------------------------------
__global__ void __launch_bounds__(256)
gat_init_kernel(float* __restrict__ out, float* __restrict__ m, float* __restrict__ denom) {
    int tid = blockIdx.x * blockDim.x + threadIdx.x;
    if (tid < N_NODES * DIM) out[tid] = 0.0f;
    if (tid < N_NODES) {
        m[tid]     = -__builtin_inff();
        denom[tid] = 0.0f;
    }
}

// ---------------------------------------------------------------------------
// Phase 1: z = h @ W via V_WMMA_F32_16X16X4_F32 (fp32, matches reference).
// Block = 8 waves; block computes a 16-row x 128-col stripe of z.
// The 16x128 A stripe (shared by all 8 waves) is staged once into padded LDS:
//   - cooperative copy: 2048 floats / 256 threads = 8 contiguous floats each
//   - row stride APAD=132 floats -> A-fragment ds_load_b64 hits distinct banks
// A layout (16x4 f32): lanes 0-15 row M=lane, v[0]=K0 v[1]=K1;
//                      lanes 16-31 same rows, v[0]=K2 v[1]=K3.
// C layout (16x16 f32): VGPR r = row r (lanes 0-15) / row r+8 (lanes 16-31).
// ---------------------------------------------------------------------------
__global__ void __launch_bounds__(256)
gat_gemm_wmma(const float* __restrict__ h, const float* __restrict__ W,
              float* __restrict__ z) {
    __shared__ float As[16 * APAD];

    const int tid     = threadIdx.y * 32 + threadIdx.x;  // 0..255
    const int rowBase = blockIdx.x * 16;                 // 3125 blocks * 16 = 50000

    // stage A tile: thread t copies 8 contiguous floats of row (t/16)
    {
        const int r = tid >> 4;          // 0..15
        const int c = (tid & 15) * 8;    // 0..120
        const float* gp = h + (size_t)(rowBase + r) * DIM + c;
        float*       lp = As + r * APAD + c;
        #pragma unroll
        for (int i = 0; i < 8; ++i) lp[i] = gp[i];
    }
    __syncthreads();

    const int lane  = threadIdx.x;           // 0..31
    const int col   = threadIdx.y * 16 + (lane & 15);
    const int khalf = (lane >> 4) * 2;       // 0 (lanes 0-15) or 2 (lanes 16-31)

    const float* ap = As + (lane & 15) * APAD + khalf;   // LDS A-fragment base
    const float* bp = W + (size_t)khalf * DIM + col;     // global B base

    v8f c = {0.f, 0.f, 0.f, 0.f, 0.f, 0.f, 0.f, 0.f};

    #pragma unroll
    for (int k = 0; k < DIM; k += 4) {
        v2f a, b;
        a.x = ap[k];
        a.y = ap[k + 1];
        b.x = bp[k * DIM];
        b.y = bp[k * DIM + DIM];
        // 8 args: (neg_a, A, neg_b, B, c_mod, C, reuse_a, reuse_b)
        c = __builtin_amdgcn_wmma_f32_16x16x4_f32(
                false, a, false, b, (short)0, c, false, false);
    }

    const int rstep = (lane >> 4) * 8;
    float* zp = z + (size_t)(rowBase + rstep) * DIM + col;
    #pragma unroll
    for (int r = 0; r < 8; ++r)
        zp[r * DIM] = c[r];
}

// ---------------------------------------------------------------------------
// Phase 2: per-node attention projections  s_i = z_i . a_src, d_i = z_i . a_dst
// (algebraic refactor of the reference's per-edge gathered dot products)
// One wave per node row; lane handles 4 features, shuffle-reduce.
// ---------------------------------------------------------------------------
__global__ void __launch_bounds__(256)
gat_sd_kernel(const float* __restrict__ z, const float* __restrict__ a_src,
              const float* __restrict__ a_dst,
              float* __restrict__ s, float* __restrict__ d) {
    const int row = blockIdx.x * 8 + threadIdx.y;
    if (row >= N_NODES) return;
    const int lane = threadIdx.x;

    const float* zp = z + (size_t)row * DIM + lane * 4;
    float ss = 0.f, dd = 0.f;
    #pragma unroll
    for (int i = 0; i < 4; ++i) {
        const float zv = zp[i];
        ss += zv * a_src[lane * 4 + i];
        dd += zv * a_dst[lane * 4 + i];
    }
    #pragma unroll
    for (int off = 16; off > 0; off >>= 1) {
        ss += __shfl_xor(ss, off, 32);
        dd += __shfl_xor(dd, off, 32);
    }
    if (lane == 0) { s[row] = ss; d[row] = dd; }
}

// ---------------------------------------------------------------------------
// Float atomic max via int-max (val>=0) / uint-min (val<0) punning; correct
// for mixed signs with memory initialized to -inf (0xFF800000).
// ---------------------------------------------------------------------------
__device__ inline void atomicMaxF(float* addr, float val) {
    if (val >= 0.0f) atomicMax((int*)addr, __float_as_int(val));
    else             atomicMin((unsigned int*)addr, __float_as_uint(val));
}

// ---------------------------------------------------------------------------
// Phase 3: e_j = leaky_relu(s[src]+d[dst]); segment max into m[dst]
// ---------------------------------------------------------------------------
__global__ void __launch_bounds__(256)
gat_edge_max_kernel(const int* __restrict__ src, const int* __restrict__ dst,
                    const float* __restrict__ s, const float* __restrict__ d,
                    float* __restrict__ e, float* __restrict__ m) {
    const int j = blockIdx.x * blockDim.x + threadIdx.x;
    if (j >= N_EDGES) return;
    float ev = s[src[j]] + d[dst[j]];
    ev = (ev >= 0.0f) ? ev : NEG_SLOPE * ev;
    e[j] = ev;
    atomicMaxF(&m[dst[j]], ev);
}

// ---------------------------------------------------------------------------
// Phase 4: ex_j = exp(e_j - m[dst]); segment sum into denom[dst]
// (every dst appearing on an edge has a finite max, so no guard needed here)
// ---------------------------------------------------------------------------
__global__ void __launch_bounds__(256)
gat_edge_exp_kernel(const int* __restrict__ dst, float* __restrict__ e,
                    const float* __restrict__ m, float* __restrict__ denom) {
    const int j = blockIdx.x * blockDim.x + threadIdx.x;
    if (j >= N_EDGES) return;
    const int dj = dst[j];
    const float ex = __expf(e[j] - m[dj]);
    e[j] = ex;
    atomicAdd(&denom[dj], ex);
}

// ---------------------------------------------------------------------------
// Phase 5: out[dst] += (ex/denom[dst]) * z[src]   (one wave per edge,
// 4 features per lane -> coalesced 512B row per edge; z & out are L2-resident)
// ---------------------------------------------------------------------------
__global__ void __launch_bounds__(256)
gat_scatter_kernel(const int* __restrict__ src, const int* __restrict__ dst,
                   const float* __restrict__ e, const float* __restrict__ denom,
                   const float* __restrict__ z, float* __restrict__ out) {
    const int j = blockIdx.x * 8 + threadIdx.y;
    if (j >= N_EDGES) return;
    const int sj = src[j];
    const int dj = dst[j];
    const float alpha = e[j] / denom[dj];
    const int f = threadIdx.x * 4;
    const float* zp = z   + (size_t)sj * DIM + f;
    float*       op = out + (size_t)dj * DIM + f;
    #pragma unroll
    for (int i = 0; i < 4; ++i)
        atomicAdd(op + i, alpha * zp[i]);
}

// ---------------------------------------------------------------------------
extern "C" void kernel_launch(void* const* d_in, const int* in_sizes, int n_in,
                              void* d_out, int out_size, void* d_ws, size_t ws_size,
                              hipStream_t stream) {
    const float* h     = (const float*)d_in[0];
    const int*   src   = (const int*)  d_in[1];
    const int*   dst   = (const int*)  d_in[2];
    const float* W     = (const float*)d_in[3];
    const float* a_src = (const float*)d_in[4];
    const float* a_dst = (const float*)d_in[5];
    float*       out   = (float*)d_out;

    // workspace layout (floats): z[N*128] | s[N] | d[N] | m[N] | denom[N] | e[E]
    float* ws    = (float*)d_ws;
    float* z     = ws;
    float* s     = z     + (size_t)N_NODES * DIM;
    float* dn    = s     + N_NODES;
    float* m     = dn    + N_NODES;
    float* denom = m     + N_NODES;
    float* e     = denom + N_NODES;

    gat_init_kernel<<<(N_NODES * DIM + 255) / 256, 256, 0, stream>>>(out, m, denom);
    gat_gemm_wmma<<<N_NODES / 16, dim3(32, 8), 0, stream>>>(h, W, z);
    gat_sd_kernel<<<(N_NODES + 7) / 8, dim3(32, 8), 0, stream>>>(z, a_src, a_dst, s, dn);
    gat_edge_max_kernel<<<(N_EDGES + 255) / 256, 256, 0, stream>>>(src, dst, s, dn, e, m);
    gat_edge_exp_kernel<<<(N_EDGES + 255) / 256, 256, 0, stream>>>(dst, e, m, denom);
    gat_scatter_kernel<<<(N_EDGES + 7) / 8, dim3(32, 8), 0, stream>>>(src, dst, e, denom, z, out);
}